// Instance_GAT_dgl_1322849927568
// MI455X (gfx1250) — compile-verified
//
#include <hip/hip_runtime.h>
#include <hip/hip_bf16.h>

typedef __attribute__((ext_vector_type(2))) float v2f;
typedef __attribute__((ext_vector_type(8))) float v8f;

#define NEG_SLOPE 0.2f
#define ORD_NEG_INF 0x007FFFFFu   // f2ord(-inf)

// ---- monotone float<->uint encoding for atomicMax on floats ----
__device__ __forceinline__ unsigned f2ord(float f) {
    unsigned u = __float_as_uint(f);
    return (u & 0x80000000u) ? ~u : (u | 0x80000000u);
}
__device__ __forceinline__ float ord2f(unsigned u) {
    return (u & 0x80000000u) ? __uint_as_float(u & 0x7FFFFFFFu)
                             : __uint_as_float(~u);
}
__device__ __forceinline__ float lrelu(float x) {
    return x > 0.0f ? x : x * NEG_SLOPE;
}

// ============================================================
// fp32 GEMM via V_WMMA_F32_16X16X4_F32 (exact fp32 numerics).
// C[M,N] = A[M,K] * B[K,N], row-major; N,K compile-time so every
// B access is base + immediate offset and the K loop fully unrolls.
// Block = 128 threads (4 waves). Each wave owns a 64x16 output strip:
// 4 M-tiles sharing one B fragment per K-step -> 4 WMMAs per B load pair.
//   blockIdx.x -> 64 rows, blockIdx.y*4 + waveid -> 16-col tile.
// A frag (16x4 f32): lane l -> row (l&15), K pair base (l>>4)*2.
// B frag (4x16 f32): lane l -> col (l&15), K pair base (l>>4)*2.
// C/D (16x16 f32):   vgpr i -> row i + (l>>4)*8, col (l&15).
// ============================================================
template <int N, int K>
__global__ void wmma_gemm_f32_t(const float* __restrict__ A,
                                const float* __restrict__ B,
                                float* __restrict__ C, int M) {
    const int lane   = threadIdx.x & 31;
    const int wid    = threadIdx.x >> 5;            // 0..3
    const int nTile  = blockIdx.y * 4 + wid;
    if (nTile * 16 >= N) return;                    // wave-uniform
    const int m0     = blockIdx.x * 64;
    const int n0     = nTile * 16;
    const int half   = lane >> 4;                   // 0 or 1
    const int lane16 = lane & 15;
    const int bcol   = n0 + lane16;

    // Per-lane base pointers: 4 A-tile rows + 1 B column; K-loop offsets
    // are compile-time immediates from here on.
    const float* Ap0;
    const float* Ap1;
    const float* Ap2;
    const float* Ap3;
    {
        int r0 = m0 + 0  + lane16; if (r0 >= M) r0 = M - 1;
        int r1 = m0 + 16 + lane16; if (r1 >= M) r1 = M - 1;
        int r2 = m0 + 32 + lane16; if (r2 >= M) r2 = M - 1;
        int r3 = m0 + 48 + lane16; if (r3 >= M) r3 = M - 1;
        Ap0 = A + (size_t)r0 * K + half * 2;
        Ap1 = A + (size_t)r1 * K + half * 2;
        Ap2 = A + (size_t)r2 * K + half * 2;
        Ap3 = A + (size_t)r3 * K + half * 2;
    }
    const float* Bp = B + (size_t)(half * 2) * N + bcol;

    v8f acc0 = {}, acc1 = {}, acc2 = {}, acc3 = {};
#pragma unroll
    for (int kk = 0; kk < K; kk += 4) {
        v2f b;
        b.x = Bp[(size_t)(kk + 0) * N];
        b.y = Bp[(size_t)(kk + 1) * N];
        const v2f a0 = *(const v2f*)(Ap0 + kk);
        const v2f a1 = *(const v2f*)(Ap1 + kk);
        const v2f a2 = *(const v2f*)(Ap2 + kk);
        const v2f a3 = *(const v2f*)(Ap3 + kk);
        acc0 = __builtin_amdgcn_wmma_f32_16x16x4_f32(false, a0, false, b, (short)0, acc0, false, false);
        acc1 = __builtin_amdgcn_wmma_f32_16x16x4_f32(false, a1, false, b, (short)0, acc1, false, false);
        acc2 = __builtin_amdgcn_wmma_f32_16x16x4_f32(false, a2, false, b, (short)0, acc2, false, false);
        acc3 = __builtin_amdgcn_wmma_f32_16x16x4_f32(false, a3, false, b, (short)0, acc3, false, false);
    }

    float* Cp = C + n0 + lane16;
#pragma unroll
    for (int i = 0; i < 8; ++i) {
        const int r = i + half * 8;
        int row;
        row = m0 + 0  + r; if (row < M) Cp[(size_t)row * N] = acc0[i];
        row = m0 + 16 + r; if (row < M) Cp[(size_t)row * N] = acc1[i];
        row = m0 + 32 + r; if (row < M) Cp[(size_t)row * N] = acc2[i];
        row = m0 + 48 + r; if (row < M) Cp[(size_t)row * N] = acc3[i];
    }
}

// ============================================================
// elementwise / edge kernels
// ============================================================
__global__ void fill_f32(float* __restrict__ p, long n, float v) {
    long i = (long)blockIdx.x * blockDim.x + threadIdx.x;
    if (i < n) p[i] = v;
}
__global__ void fill_u32(unsigned* __restrict__ p, long n, unsigned v) {
    long i = (long)blockIdx.x * blockDim.x + threadIdx.x;
    if (i < n) p[i] = v;
}

// el[n*heads], er[n*heads] from feat [n, heads, d] and attn vectors [heads, d]
template <int HEADS, int D>
__global__ void el_er_kernel(const float* __restrict__ feat,
                             const float* __restrict__ al,
                             const float* __restrict__ ar,
                             float* __restrict__ el, float* __restrict__ er,
                             int n) {
    const int idx = blockIdx.x * blockDim.x + threadIdx.x;   // node*HEADS + h
    if (idx >= n * HEADS) return;
    const int h = idx % HEADS;
    const float* f = feat + (size_t)idx * D;
    float sl = 0.0f, sr = 0.0f;
#pragma unroll 8
    for (int i = 0; i < D; ++i) {
        const float v = f[i];
        sl += v * al[h * D + i];
        sr += v * ar[h * D + i];
    }
    el[idx] = sl;
    er[idx] = sr;
}

// pass 1: segment max of leaky_relu(el[src]+er[dst]) into mord[dst]
template <int HEADS>
__global__ void edge_max(const int* __restrict__ src, const int* __restrict__ dst,
                         const float* __restrict__ el, const float* __restrict__ er,
                         unsigned* __restrict__ mord, int E) {
    long idx = (long)blockIdx.x * blockDim.x + threadIdx.x;
    if (idx >= (long)E * HEADS) return;
    const int e = (int)(idx / HEADS), h = (int)(idx % HEADS);
    const int s = src[e], t = dst[e];
    const float x = lrelu(el[s * HEADS + h] + er[t * HEADS + h]);
    atomicMax(&mord[t * HEADS + h], f2ord(x));
}

// pass 2: ex = exp(x - m[dst]); store numerator, atomic-sum into ssum[dst]
template <int HEADS>
__global__ void edge_expsum(const int* __restrict__ src, const int* __restrict__ dst,
                            const float* __restrict__ el, const float* __restrict__ er,
                            const unsigned* __restrict__ mord,
                            float* __restrict__ alpha, float* __restrict__ ssum,
                            int E) {
    long idx = (long)blockIdx.x * blockDim.x + threadIdx.x;
    if (idx >= (long)E * HEADS) return;
    const int e = (int)(idx / HEADS), h = (int)(idx % HEADS);
    const int s = src[e], t = dst[e];
    const float x = lrelu(el[s * HEADS + h] + er[t * HEADS + h]);
    const float m = ord2f(mord[t * HEADS + h]);
    const float ex = __expf(x - m);
    alpha[idx] = ex;
    atomicAdd(&ssum[t * HEADS + h], ex);
}

// pass 3: normalize alpha
template <int HEADS>
__global__ void alpha_norm(const int* __restrict__ dst, float* __restrict__ alpha,
                           const float* __restrict__ ssum, int E) {
    long idx = (long)blockIdx.x * blockDim.x + threadIdx.x;
    if (idx >= (long)E * HEADS) return;
    const int e = (int)(idx / HEADS), h = (int)(idx % HEADS);
    alpha[idx] = alpha[idx] / ssum[dst[e] * HEADS + h];
}

// aggregation: rst[dst, h, :] += feat[src, h, :] * alpha[e, h]
// 64 threads per edge (D fixed at 64), 4 edges per 256-thread block.
// Gathers/atomics are L2-resident (feat table fits in the 192MB L2).
template <int HEADS>
__global__ void aggregate64(const int* __restrict__ src, const int* __restrict__ dst,
                            const float* __restrict__ feat,
                            const float* __restrict__ alpha,
                            float* __restrict__ rst, int E) {
    const int e = blockIdx.x * 4 + (threadIdx.x >> 6);
    if (e >= E) return;
    const int i = threadIdx.x & 63;
    const int s = src[e], t = dst[e];
    const float* fs = feat + (size_t)s * (HEADS * 64) + i;
    float*       rt = rst  + (size_t)t * (HEADS * 64) + i;
#pragma unroll
    for (int h = 0; h < HEADS; ++h) {
        const float a = alpha[(size_t)e * HEADS + h];
        atomicAdd(rt + h * 64, fs[h * 64] * a);
    }
}

// h = relu(rst + bias)
__global__ void relu_bias(const float* __restrict__ rst, const float* __restrict__ b,
                          float* __restrict__ h, long n, int c) {
    long idx = (long)blockIdx.x * blockDim.x + threadIdx.x;
    if (idx >= n) return;
    const float v = rst[idx] + b[idx % c];
    h[idx] = v > 0.0f ? v : 0.0f;
}

// out[node, c] = bias[c]  (layer-2 aggregation then adds into it)
__global__ void out_bias(float* __restrict__ out, const float* __restrict__ b,
                         long n, int c) {
    long idx = (long)blockIdx.x * blockDim.x + threadIdx.x;
    if (idx >= n) return;
    out[idx] = b[idx % c];
}

// ============================================================
// orchestration
// ============================================================
static inline dim3 g1d(long n, int blk = 256) {
    return dim3((unsigned)((n + blk - 1) / blk));
}

extern "C" void kernel_launch(void* const* d_in, const int* in_sizes, int n_in,
                              void* d_out, int out_size, void* d_ws, size_t ws_size,
                              hipStream_t stream) {
    const float* emb = (const float*)d_in[0];
    const int*   src = (const int*)d_in[1];
    const int*   dst = (const int*)d_in[2];
    const float* W1  = (const float*)d_in[3];
    const float* al1 = (const float*)d_in[4];
    const float* ar1 = (const float*)d_in[5];
    const float* b1  = (const float*)d_in[6];
    const float* W2  = (const float*)d_in[7];
    const float* al2 = (const float*)d_in[8];
    const float* ar2 = (const float*)d_in[9];
    const float* b2  = (const float*)d_in[10];
    float* out = (float*)d_out;

    constexpr int IN = 128, H1 = 4, D1 = 64, C1 = H1 * D1 /*256*/, D2 = 64;
    const int N = in_sizes[0] / IN;       // 50000
    const int E = in_sizes[1];            // 800000

    // ---- workspace layout (floats) ----
    float* ws = (float*)d_ws;
    size_t off = 0;
    float*    feat1  = ws + off; off += (size_t)N * C1;   // layer1 feat, later h
    float*    rst1   = ws + off; off += (size_t)N * C1;   // layer1 agg, later feat2
    float*    el1    = ws + off; off += (size_t)N * H1;
    float*    er1    = ws + off; off += (size_t)N * H1;
    float*    s1     = ws + off; off += (size_t)N * H1;
    unsigned* m1     = (unsigned*)(ws + off); off += (size_t)N * H1;
    float*    alpha1 = ws + off; off += (size_t)E * H1;   // reused as alpha2
    float*    el2    = ws + off; off += N;
    float*    er2    = ws + off; off += N;
    float*    s2     = ws + off; off += N;
    unsigned* m2     = (unsigned*)(ws + off); off += N;
    float*    feat2  = rst1;      // rst1 free after relu_bias
    float*    alpha2 = alpha1;    // alpha1 free after layer1 aggregation

    const dim3 blk(256);

    // ===================== layer 1 =====================
    fill_f32<<<g1d((long)N * C1), blk, 0, stream>>>(rst1, (long)N * C1, 0.0f);
    fill_f32<<<g1d((long)N * H1), blk, 0, stream>>>(s1, (long)N * H1, 0.0f);
    fill_u32<<<g1d((long)N * H1), blk, 0, stream>>>(m1, (long)N * H1, ORD_NEG_INF);

    // feat1 = emb @ W1   [N,128]x[128,256]
    wmma_gemm_f32_t<C1, IN><<<dim3((N + 63) / 64, C1 / 64), dim3(128), 0, stream>>>(
        emb, W1, feat1, N);

    el_er_kernel<H1, D1><<<g1d((long)N * H1), blk, 0, stream>>>(feat1, al1, ar1, el1, er1, N);
    edge_max<H1>   <<<g1d((long)E * H1), blk, 0, stream>>>(src, dst, el1, er1, m1, E);
    edge_expsum<H1><<<g1d((long)E * H1), blk, 0, stream>>>(src, dst, el1, er1, m1, alpha1, s1, E);
    alpha_norm<H1> <<<g1d((long)E * H1), blk, 0, stream>>>(dst, alpha1, s1, E);
    aggregate64<H1><<<dim3((E + 3) / 4), blk, 0, stream>>>(src, dst, feat1, alpha1, rst1, E);

    // h = relu(rst1 + b1) -> feat1 buffer
    relu_bias<<<g1d((long)N * C1), blk, 0, stream>>>(rst1, b1, feat1, (long)N * C1, C1);

    // ===================== layer 2 =====================
    fill_f32<<<g1d(N), blk, 0, stream>>>(s2, N, 0.0f);
    fill_u32<<<g1d(N), blk, 0, stream>>>(m2, N, ORD_NEG_INF);

    // feat2 = h @ W2   [N,256]x[256,64]
    wmma_gemm_f32_t<D2, C1><<<dim3((N + 63) / 64, D2 / 64), dim3(128), 0, stream>>>(
        feat1, W2, feat2, N);

    el_er_kernel<1, D2><<<g1d(N), blk, 0, stream>>>(feat2, al2, ar2, el2, er2, N);
    edge_max<1>   <<<g1d(E), blk, 0, stream>>>(src, dst, el2, er2, m2, E);
    edge_expsum<1><<<g1d(E), blk, 0, stream>>>(src, dst, el2, er2, m2, alpha2, s2, E);
    alpha_norm<1> <<<g1d(E), blk, 0, stream>>>(dst, alpha2, s2, E);

    // out = b2 broadcast, then scatter-add messages
    out_bias<<<g1d((long)N * D2), blk, 0, stream>>>(out, b2, (long)N * D2, D2);
    aggregate64<1><<<dim3((E + 3) / 4), blk, 0, stream>>>(src, dst, feat2, alpha2, out, E);
}